// NetSm_74637941669908
// MI455X (gfx1250) — compile-verified
//
#include <hip/hip_runtime.h>

// fp32 WMMA fragment types for V_WMMA_F32_16X16X4_F32 (wave32):
//   A: 16x4 fp32 -> 2 VGPRs/lane, B: 4x16 fp32 -> 2 VGPRs/lane, C/D: 16x16 fp32 -> 8 VGPRs/lane
typedef __attribute__((ext_vector_type(2))) float v2f;
typedef __attribute__((ext_vector_type(8))) float v8f;

#define KVOL 27

// One wave computes an MT*16-voxel x COUT output block:
//   out[MT*16, COUT] = sum_k gather(feat, nbr[:,k])[MT*16, CIN] @ w[k][CIN, COUT]
// K is split into ceil(CIN/4) subtiles for V_WMMA_F32_16X16X4_F32; columns into
// ceil(COUT/16) 16-wide tiles (zero-padded when COUT<16). Each B (weight)
// fragment is reused across the MT row tiles, cutting weight vmem traffic MT x.
//
// A layout (16x4 f32, ISA 7.12.2): lanes 0-15 hold M=lane, {K=0,K=1};
// lanes 16-31 hold M=lane-16, {K=2,K=3}.
// B layout (4x16 f32): lanes 0-15 hold N=lane rows {K=0,K=1};
// lanes 16-31 hold N=lane-16 rows {K=2,K=3}.
// C/D layout (16x16 f32): VGPR r, lanes 0-15 -> M=r, N=lane; lanes 16-31 -> M=r+8.
template <int CIN, int COUT, int MT>
__global__ __launch_bounds__(256) void subm_conv_wmma(
    const float* __restrict__ feat,   // (N, CIN)
    const float* __restrict__ w,      // (KVOL, CIN, COUT)
    const int* __restrict__ nbr,      // (N, KVOL), sentinel == N
    float* __restrict__ out,          // (N, COUT)
    int n) {
  constexpr int NDT = (COUT + 15) / 16;  // 16-wide output column tiles (padded)
  constexpr int NKT = (CIN + 3) / 4;     // K subtiles of 4 input channels
  constexpr bool COL_FULL = (COUT % 16) == 0;

  const int lane   = threadIdx.x & 31;
  const int wave   = threadIdx.x >> 5;
  const int block0 = (blockIdx.x * (blockDim.x >> 5) + wave) * (MT * 16);
  if (block0 >= n) return;               // wave-uniform exit

  const int mlane = lane & 15;           // voxel-in-tile / output column
  const int khalf = lane >> 4;           // 0: K pair {0,1}, 1: K pair {2,3}

  v8f acc[MT][NDT];
#pragma unroll
  for (int mt = 0; mt < MT; ++mt)
#pragma unroll
    for (int t = 0; t < NDT; ++t)
      acc[mt][t] = (v8f){0.f, 0.f, 0.f, 0.f, 0.f, 0.f, 0.f, 0.f};

  for (int k = 0; k < KVOL; ++k) {
    // Gather neighbor row indices for each row tile (sentinel n -> zero row)
    int r[MT];
#pragma unroll
    for (int mt = 0; mt < MT; ++mt) {
      const int row0 = block0 + mt * 16;     // wave-uniform
      if (row0 < n) {                        // uniform branch
        const int voxel = row0 + mlane;
        r[mt] = (voxel < n) ? nbr[voxel * KVOL + k] : n;
      } else {
        r[mt] = n;
      }
    }
    const float* __restrict__ wk = w + (size_t)k * CIN * COUT;

#pragma unroll
    for (int kt = 0; kt < NKT; ++kt) {
      const int c0 = kt * 4 + khalf * 2;     // this lane's channel pair

      // A fragments for all MT row tiles at this K subtile
      v2f a[MT];
#pragma unroll
      for (int mt = 0; mt < MT; ++mt) {
        a[mt] = (v2f){0.f, 0.f};
        if (r[mt] < n) {
          const float* __restrict__ fr = feat + (size_t)r[mt] * CIN;
          if constexpr ((CIN & 3) == 0) {
            a[mt] = *(const v2f*)(fr + c0);  // 8B-aligned pair load
          } else {
            if (c0 < CIN)     a[mt].x = fr[c0];
            if (c0 + 1 < CIN) a[mt].y = fr[c0 + 1];
          }
        }
      }

#pragma unroll
      for (int t = 0; t < NDT; ++t) {
        const int col = t * 16 + mlane;
        const bool colOK = COL_FULL || (col < COUT);
        v2f b = (v2f){0.f, 0.f};
        if (colOK) {
          if (c0 < CIN)     b.x = wk[c0 * COUT + col];
          if (c0 + 1 < CIN) b.y = wk[(c0 + 1) * COUT + col];
        }
        // One B fragment feeds MT WMMAs (distinct accumulators -> no RAW chain)
#pragma unroll
        for (int mt = 0; mt < MT; ++mt) {
          acc[mt][t] = __builtin_amdgcn_wmma_f32_16x16x4_f32(
              false, a[mt], false, b, (short)0, acc[mt][t], false, false);
        }
      }
    }
  }

  // Write back D: lane<16 -> rows +r, lane>=16 -> rows +r+8, col = t*16 + mlane
#pragma unroll
  for (int mt = 0; mt < MT; ++mt) {
    const int row0 = block0 + mt * 16;
    if (row0 >= n) break;                    // uniform
#pragma unroll
    for (int t = 0; t < NDT; ++t) {
      const int col = t * 16 + mlane;
      const bool colOK = COL_FULL || (col < COUT);
      if (colOK) {
#pragma unroll
        for (int rr = 0; rr < 8; ++rr) {
          const int m = row0 + rr + khalf * 8;
          if (m < n) out[(size_t)m * COUT + col] = acc[mt][t][rr];
        }
      }
    }
  }
}

extern "C" void kernel_launch(void* const* d_in, const int* in_sizes, int n_in,
                              void* d_out, int out_size, void* d_ws, size_t ws_size,
                              hipStream_t stream) {
  const float* feat = (const float*)d_in[0];   // (N, 3)
  const float* w0   = (const float*)d_in[1];   // (27, 3, 8)
  const float* w1   = (const float*)d_in[2];   // (27, 8, 16)
  const float* w2   = (const float*)d_in[3];   // (27, 16, 32)
  const float* w3   = (const float*)d_in[4];   // (27, 32, 64)
  const int*   nbr  = (const int*)d_in[5];     // (N, 27)
  float* out = (float*)d_out;                  // (N, 64)

  const int n = in_sizes[0] / 3;               // N = 150000

  // Workspace: intermediate activations (fp32)
  float* x1 = (float*)d_ws;                    // (N, 8)
  float* x2 = x1 + (size_t)n * 8;              // (N, 16)
  float* x3 = x2 + (size_t)n * 16;             // (N, 32)

  constexpr int MT = 4;                        // row tiles (of 16) per wave
  const int wavesPerBlock = 8;                 // 256 threads = 8 wave32
  const int rowsPerBlock = wavesPerBlock * MT * 16;
  const dim3 block(256);
  const dim3 grid((n + rowsPerBlock - 1) / rowsPerBlock);

  subm_conv_wmma<3, 8, MT>  <<<grid, block, 0, stream>>>(feat, w0, nbr, x1,  n);
  subm_conv_wmma<8, 16, MT> <<<grid, block, 0, stream>>>(x1,   w1, nbr, x2,  n);
  subm_conv_wmma<16, 32, MT><<<grid, block, 0, stream>>>(x2,   w2, nbr, x3,  n);
  subm_conv_wmma<32, 64, MT><<<grid, block, 0, stream>>>(x3,   w3, nbr, out, n);
}